// VectorQuantizer_1743756722217
// MI455X (gfx1250) — compile-verified
//
#include <hip/hip_runtime.h>

#define N_TOKENS  16384
#define EMBED_DIM 256
#define N_CODES   8192
#define ZSTRIDE   260   // 256 + 4 pad -> LDS bank-conflict-free row access
#define MTILES    2     // M-tiles per wave: B fragments reused across both
#define ROWS_PER_BLOCK (16 * MTILES)   // 32 tokens per workgroup

typedef __attribute__((ext_vector_type(2))) float v2f;
typedef __attribute__((ext_vector_type(8))) float v8f;

// ---------------------------------------------------------------------------
// Kernel 1: enorm[k] = ||e_k||^2. One wave per code row (coalesced float4).
// ---------------------------------------------------------------------------
__global__ void vq_enorm_kernel(const float* __restrict__ e,
                                float* __restrict__ enorm) {
  const int code = (blockIdx.x * blockDim.x + threadIdx.x) >> 5;
  const int lane = threadIdx.x & 31;
  if (code >= N_CODES) return;
  const float4* row = (const float4*)(e + (size_t)code * EMBED_DIM);
  float4 a = row[lane];        // 64 float4 per row, 32 lanes x 2
  float4 b = row[lane + 32];
  float s = a.x * a.x + a.y * a.y + a.z * a.z + a.w * a.w +
            b.x * b.x + b.y * b.y + b.z * b.z + b.w * b.w;
  #pragma unroll
  for (int off = 16; off > 0; off >>= 1) s += __shfl_xor(s, off, 32);
  if (lane == 0) enorm[code] = s;
}

// ---------------------------------------------------------------------------
// Kernel 2: fused Z·E^T (f32 WMMA) + running argmin + gather epilogue.
// One workgroup (8 waves) per 32-token tile. Each wave owns a disjoint
// interleaved 16-code column slice per iteration (64 iters cover all 8192
// codes) and computes TWO 16x16 WMMA tiles per B fragment, so every
// global_load_b64 of the codebook feeds 2 wmma ops (L2 traffic halved).
// ---------------------------------------------------------------------------
__global__ __launch_bounds__(256, 2) void vq_argmin_kernel(
    const float* __restrict__ z, const float* __restrict__ e,
    const float* __restrict__ enorm, float* __restrict__ outq,
    float* __restrict__ outidx) {
  __shared__ float zs[ROWS_PER_BLOCK * ZSTRIDE];
  __shared__ float redval[8 * ROWS_PER_BLOCK];
  __shared__ int   redidx[8 * ROWS_PER_BLOCK];
  __shared__ int   finalidx[ROWS_PER_BLOCK];

  const int tid  = threadIdx.x;
  const int wave = tid >> 5;
  const int lane = tid & 31;
  const int row0 = blockIdx.x * ROWS_PER_BLOCK;

  // Stage the 32x256 Z tile into LDS (coalesced: iteration i = row i, col tid).
  #pragma unroll
  for (int i = 0; i < ROWS_PER_BLOCK; ++i)
    zs[i * ZSTRIDE + tid] = z[(size_t)(row0 + i) * EMBED_DIM + tid];
  __syncthreads();

  // WMMA f32 16x16x4 operand lane mapping:
  //   A (16x4): lanes 0-15 -> M=lane, K={0,1}; lanes 16-31 -> M=lane-16, K={2,3}
  //   B (4x16): lanes 0-15 -> N=lane, K={0,1}; lanes 16-31 -> N=lane-16, K={2,3}
  const int laneRow = lane & 15;
  const int hi      = lane >> 4;
  const int kOff    = 2 * hi;
  const float* zrow0 = zs + laneRow * ZSTRIDE + kOff;            // M-tile 0
  const float* zrow1 = zrow0 + 16 * ZSTRIDE;                     // M-tile 1

  float bestval[MTILES][8];
  int   bestidx[MTILES][8];
  #pragma unroll
  for (int m = 0; m < MTILES; ++m)
    #pragma unroll
    for (int r = 0; r < 8; ++r) {
      bestval[m][r] = __builtin_inff();
      bestidx[m][r] = 0;
    }

  for (int t = 0; t < N_CODES / (8 * 16); ++t) {
    const int code = (t * 8 + wave) * 16 + laneRow;
    const float* erow = e + (size_t)code * EMBED_DIM + kOff;     // L2-resident
    v8f acc0 = {};
    v8f acc1 = {};
    #pragma unroll 8
    for (int kk = 0; kk < EMBED_DIM / 4; ++kk) {
      v2f b  = *(const v2f*)(erow + 4 * kk);   // global_load_b64 (shared by 2 wmma)
      v2f a0 = *(const v2f*)(zrow0 + 4 * kk);  // ds_load_b64, conflict-free
      v2f a1 = *(const v2f*)(zrow1 + 4 * kk);
      acc0 = __builtin_amdgcn_wmma_f32_16x16x4_f32(
          false, a0, false, b, (short)0, acc0, false, false);
      acc1 = __builtin_amdgcn_wmma_f32_16x16x4_f32(
          false, a1, false, b, (short)0, acc1, false, false);
    }
    // score_k = ||e_k||^2 - 2 z.e_k   (||z||^2 dropped: constant per token)
    const float en = enorm[code];
    #pragma unroll
    for (int r = 0; r < 8; ++r) {
      float s0 = __builtin_fmaf(-2.0f, acc0[r], en);
      if (s0 < bestval[0][r]) { bestval[0][r] = s0; bestidx[0][r] = code; }
      float s1 = __builtin_fmaf(-2.0f, acc1[r], en);
      if (s1 < bestval[1][r]) { bestval[1][r] = s1; bestidx[1][r] = code; }
    }
  }

  // Min-reduce (val,idx) across the 16 lanes of each half-group.
  // C/D slot r of tile m holds token row (16*m + r + 8*hi); column = laneRow.
  #pragma unroll
  for (int off = 8; off > 0; off >>= 1) {
    #pragma unroll
    for (int m = 0; m < MTILES; ++m)
      #pragma unroll
      for (int r = 0; r < 8; ++r) {
        float ov = __shfl_xor(bestval[m][r], off, 32);
        int   oi = __shfl_xor(bestidx[m][r], off, 32);
        if (ov < bestval[m][r] ||
            (ov == bestval[m][r] && oi < bestidx[m][r])) {
          bestval[m][r] = ov; bestidx[m][r] = oi;
        }
      }
  }
  if (laneRow == 0) {
    #pragma unroll
    for (int m = 0; m < MTILES; ++m)
      #pragma unroll
      for (int r = 0; r < 8; ++r) {
        const int row = 16 * m + r + 8 * hi;
        redval[wave * ROWS_PER_BLOCK + row] = bestval[m][r];
        redidx[wave * ROWS_PER_BLOCK + row] = bestidx[m][r];
      }
  }
  __syncthreads();

  // Cross-wave reduce: thread i owns token row i (first-min tie-break).
  if (tid < ROWS_PER_BLOCK) {
    float bv = redval[tid];
    int   bi = redidx[tid];
    #pragma unroll
    for (int w = 1; w < 8; ++w) {
      float v = redval[w * ROWS_PER_BLOCK + tid];
      int   i = redidx[w * ROWS_PER_BLOCK + tid];
      if (v < bv || (v == bv && i < bi)) { bv = v; bi = i; }
    }
    finalidx[tid] = bi;
    outidx[row0 + tid] = (float)bi;   // exact: idx < 8192
  }
  __syncthreads();

  // Gather quantized = e[argmin]; coalesced 256-wide row copies.
  #pragma unroll
  for (int i = 0; i < ROWS_PER_BLOCK; ++i) {
    const int ci = finalidx[i];
    outq[(size_t)(row0 + i) * EMBED_DIM + tid] =
        e[(size_t)ci * EMBED_DIM + tid];
  }
}

// ---------------------------------------------------------------------------
extern "C" void kernel_launch(void* const* d_in, const int* in_sizes, int n_in,
                              void* d_out, int out_size, void* d_ws,
                              size_t ws_size, hipStream_t stream) {
  (void)in_sizes; (void)n_in; (void)out_size; (void)ws_size;
  const float* z = (const float*)d_in[0];            // [16384, 256] f32
  const float* e = (const float*)d_in[1];            // [8192, 256]  f32
  float* outq   = (float*)d_out;                     // [16384*256] quantized
  float* outidx = (float*)d_out + (size_t)N_TOKENS * EMBED_DIM;  // [16384] idx
  float* enorm  = (float*)d_ws;                      // 8192 floats scratch

  vq_enorm_kernel<<<(N_CODES * 32) / 256, 256, 0, stream>>>(e, enorm);
  vq_argmin_kernel<<<N_TOKENS / ROWS_PER_BLOCK, 256, 0, stream>>>(
      z, e, enorm, outq, outidx);
}